// GatherPaKvCacheInplaceModel_17111149707758
// MI455X (gfx1250) — compile-verified
//
#include <hip/hip_runtime.h>

// Problem constants (from the reference):
//   key_cache/value_cache : (2048, 128, 8, 128) fp16
//   block_tables          : (32, 32) int32
//   seq_lens              : (32,) int32
//   key/value             : (32, 4096, 8, 128) fp16
#define NUM_BLOCKS 2048
#define BLOCK_SIZE 128
#define N_HEADS    8
#define HEAD_DIM   128
#define BATCH      32
#define MAX_SEQ    4096
#define BT_COLS    (MAX_SEQ / BLOCK_SIZE)            // 32
#define ROW_VECS   (N_HEADS * HEAD_DIM / 8)          // 128 x 16B vectors per (b,s)
#define TOTAL_VECS (BATCH * MAX_SEQ * ROW_VECS)      // 16,777,216 per tensor

// 16-byte vector = 8 fp16 elements; ext_vector so nontemporal builtins accept it.
typedef __attribute__((ext_vector_type(4))) unsigned int uint32x4;

__global__ __launch_bounds__(128) void gather_pa_kv_kernel(
    const uint32x4* __restrict__ key_cache,
    const uint32x4* __restrict__ value_cache,
    const int*      __restrict__ block_tables,
    const int*      __restrict__ seq_lens,
    const uint32x4* __restrict__ key_in,
    const uint32x4* __restrict__ value_in,
    uint32x4*       __restrict__ key_out,
    uint32x4*       __restrict__ value_out,
    const int*      __restrict__ p_seq_offset,
    const int*      __restrict__ p_is_cumsum)
{
    // One block == one (b, s) row of 8 heads x 128 dims (2 KB of K + 2 KB of V).
    // Everything derived from blockIdx is wave-uniform -> scalar loads/branches.
    const int bs = blockIdx.x;              // b * MAX_SEQ + s
    const int b  = bs >> 12;                // / 4096
    const int s  = bs & (MAX_SEQ - 1);
    const int t  = threadIdx.x;             // 0..127: h = t>>4, d8 = t&15

    const int seq_offset = p_seq_offset[0]; // uniform -> s_load
    const int is_cumsum  = p_is_cumsum[0];

    const int len = is_cumsum ? (seq_lens[b + 1] - seq_lens[b]) : seq_lens[b];

    // Flat 16B-vector index into key/value/out layout (b, s, h, d): < 2^24.
    const int dstV = bs * ROW_VECS + t;

    if (s < len) {
        // Gather path: logical position -> physical cache block.
        const int pos  = s + seq_offset;
        const int slot = pos & (BLOCK_SIZE - 1);     // mod matches Python semantics
        int blk = pos >> 7;                          // floor-div 128 (arith shift)
        blk = blk < 0 ? 0 : (blk > BT_COLS - 1 ? BT_COLS - 1 : blk);  // JAX-style clamp
        const int phys = block_tables[b * BT_COLS + blk];             // uniform -> s_load

        // Cache vector index: ((phys*128 + slot)*8 + h)*128/8 + ... : < 2^25.
        const int srcV = (phys * BLOCK_SIZE + slot) * ROW_VECS + t;

        const uint32x4 rk = __builtin_nontemporal_load(&key_cache[srcV]);
        const uint32x4 rv = __builtin_nontemporal_load(&value_cache[srcV]);
        __builtin_nontemporal_store(rk, &key_out[dstV]);
        __builtin_nontemporal_store(rv, &value_out[dstV]);
    } else {
        // Pass-through path: copy the user-provided key/value.
        const uint32x4 rk = __builtin_nontemporal_load(&key_in[dstV]);
        const uint32x4 rv = __builtin_nontemporal_load(&value_in[dstV]);
        __builtin_nontemporal_store(rk, &key_out[dstV]);
        __builtin_nontemporal_store(rv, &value_out[dstV]);
    }
}

extern "C" void kernel_launch(void* const* d_in, const int* in_sizes, int n_in,
                              void* d_out, int out_size, void* d_ws, size_t ws_size,
                              hipStream_t stream) {
    (void)in_sizes; (void)n_in; (void)out_size; (void)d_ws; (void)ws_size;

    const uint32x4* key_cache   = (const uint32x4*)d_in[0];
    const uint32x4* value_cache = (const uint32x4*)d_in[1];
    const int*      block_tab   = (const int*)d_in[2];
    const int*      seq_lens    = (const int*)d_in[3];
    const uint32x4* key_in      = (const uint32x4*)d_in[4];
    const uint32x4* value_in    = (const uint32x4*)d_in[5];
    const int*      seq_offset  = (const int*)d_in[6];
    const int*      is_cumsum   = (const int*)d_in[7];

    // d_out = key_out (flat) ++ value_out (flat), fp16.
    uint32x4* key_out   = (uint32x4*)d_out;
    uint32x4* value_out = key_out + TOTAL_VECS;

    const int grid = BATCH * MAX_SEQ;   // 131072 blocks, one (b,s) row each
    gather_pa_kv_kernel<<<grid, 128, 0, stream>>>(
        key_cache, value_cache, block_tab, seq_lens,
        key_in, value_in, key_out, value_out,
        seq_offset, is_cumsum);
}